// ACMModule_15290083574329
// MI455X (gfx1250) — compile-verified
//
#include <hip/hip_runtime.h>
#include <hip/hip_bf16.h>

#define DIM 256

typedef float v2f __attribute__((ext_vector_type(2)));
typedef float v8f __attribute__((ext_vector_type(8)));
typedef unsigned int u32x4 __attribute__((ext_vector_type(4)));
typedef int i32x4 __attribute__((ext_vector_type(4)));
typedef int i32x8 __attribute__((ext_vector_type(8)));

// ---------------------------------------------------------------------------
// TDM: issue a 2-D tensor_load_to_lds (fp32 elements).
// D# group0: count=1 | lds_addr | global_addr[56:0] | type=2
// D# group1: data_size=2 (4B), tensor_dim0/1, tile_dim0/1, dim0 stride
// Groups 2/3 zero (2-D tensor). 6-arg (clang-23) builtin form.
// ---------------------------------------------------------------------------
__device__ __forceinline__ void tdm_load_2d(
    unsigned int lds_off_bytes, const void* gptr,
    unsigned int tensor_d0, unsigned int tensor_d1,
    unsigned int tile_d0, unsigned int tile_d1,
    unsigned int stride0_elems)
{
    unsigned long long ga = (unsigned long long)gptr;
    u32x4 g0;
    g0.x = 1u;                                    // count=1, user mode
    g0.y = lds_off_bytes;                         // lds_addr
    g0.z = (unsigned int)(ga & 0xFFFFFFFFu);      // global_addr[31:0]
    g0.w = (unsigned int)((ga >> 32) & 0x1FFFFFFu) | (2u << 30); // [56:32]|type=2

    i32x8 g1;
    g1[0] = 0x00020000;                           // data_size=2 (4 bytes)
    g1[1] = (int)((tensor_d0 & 0xFFFFu) << 16);   // tensor_dim0[15:0] @ bits63:48
    g1[2] = (int)(((tensor_d0 >> 16) & 0xFFFFu) | ((tensor_d1 & 0xFFFFu) << 16));
    g1[3] = (int)(((tensor_d1 >> 16) & 0xFFFFu) | ((tile_d0 & 0xFFFFu) << 16));
    g1[4] = (int)(tile_d1 & 0xFFFFu);             // tile_dim1 (tile_dim2=0)
    g1[5] = (int)stride0_elems;                   // tensor_dim0_stride[31:0]
    g1[6] = 0;
    g1[7] = 0;

    i32x4 z4 = {0, 0, 0, 0};
    i32x8 z8 = {0, 0, 0, 0, 0, 0, 0, 0};
    __builtin_amdgcn_tensor_load_to_lds(g0, g1, z4, z4, z8, 0);
}

// ---------------------------------------------------------------------------
// GEMM: Y[N,256] = X[N,256] @ W[256,256]   (fp32, WMMA 16x16x4)
// Block = 256 threads (8 waves), covers 64 rows x 256 cols.
// TDM stages all of W (256KB) + a 64x256 X panel (64KB) into LDS (=320KB/WGP),
// then the K-loop is pure LDS + WMMA (no VMEM in the hot loop).
// Wave w: mtile = w>>1 (16 rows), ngrp = w&1 (128 cols -> 8 accum tiles).
// A-frag: lane L holds row M=L%16, K = 2*(L/16)+{0,1}  (ds_load_b64)
// B-frag: lane L holds col N=L%16, K = 2*(L/16)+{0,1}
// C/D:    vgpr r, lane L -> M = r + 8*(L/16), N = L%16
// ---------------------------------------------------------------------------
__global__ __launch_bounds__(256) void gemm_wmma_f32(
    const float* __restrict__ X, const float* __restrict__ W,
    float* __restrict__ Y, int nrows)
{
    extern __shared__ float lds[];
    float* __restrict__ wlds = lds;                 // [256][256] = 256KB
    float* __restrict__ xlds = lds + DIM * DIM;     // [64][256]  = 64KB

    const int lane  = threadIdx.x & 31;
    const int wid   = threadIdx.x >> 5;
    const int mtile = wid >> 1;                     // 0..3
    const int ngrp  = wid & 1;                      // 0..1 (128 cols)
    const int block_m0 = blockIdx.x * 64;

    // Wave 0 stages W and the X panel via the Tensor Data Mover.
    if (wid == 0) {
        tdm_load_2d((unsigned int)(size_t)wlds, W, DIM, DIM, DIM, DIM, DIM);
        unsigned int rows_left = (unsigned int)(nrows - block_m0); // OOB rows read 0
        tdm_load_2d((unsigned int)(size_t)xlds, X + (size_t)block_m0 * DIM,
                    DIM, rows_left, DIM, 64, DIM);
        __builtin_amdgcn_s_wait_tensorcnt(0);
    }
    __syncthreads();

    const int arow  = mtile * 16 + (lane & 15);     // row within x panel
    const int khalf = (lane >> 4) * 2;              // 0 or 2
    const int ncol  = lane & 15;
    const int n0    = ngrp * 128;

    v8f c[8];
    #pragma unroll
    for (int t = 0; t < 8; ++t) c[t] = (v8f){};

    const float* __restrict__ abase = xlds + arow * DIM + khalf;

    for (int k = 0; k < DIM; k += 4) {
        const v2f a = *(const v2f*)(abase + k);
        const float* __restrict__ w0 = wlds + (k + khalf) * DIM + n0 + ncol;
        const float* __restrict__ w1 = w0 + DIM;
        #pragma unroll
        for (int t = 0; t < 8; ++t) {
            v2f b;
            b.x = w0[t * 16];
            b.y = w1[t * 16];
            c[t] = __builtin_amdgcn_wmma_f32_16x16x4_f32(
                false, a, false, b, (short)0, c[t], false, false);
        }
    }

    // Store (guard tail rows only; padded rows computed on zeros)
    const int mbase = block_m0 + mtile * 16 + 8 * (lane >> 4);
    #pragma unroll
    for (int r = 0; r < 8; ++r) {
        const int grow = mbase + r;
        if (grow < nrows) {
            float* __restrict__ yrow = Y + (size_t)grow * DIM + n0 + ncol;
            #pragma unroll
            for (int t = 0; t < 8; ++t) yrow[t * 16] = c[t][r];
        }
    }
}

// ---------------------------------------------------------------------------
// SpMM scatter: acc[rows[e], :] += vals[e] * X[cols[e], :]
// One wave per edge, 8 floats per lane (2x float4 load, 8 HW fp32 atomics).
// Edge lists are stream-once -> non-temporal loads (keep L2 for XW + acc).
// X (102MB) fits in the 192MB L2 -> gathers mostly hit L2.
// unsafeAtomicAdd lowers to global_atomic_add_f32 (no CAS loop).
// ---------------------------------------------------------------------------
__global__ __launch_bounds__(256) void spmm_scatter(
    const int* __restrict__ rows, const int* __restrict__ cols,
    const float* __restrict__ vals, const float* __restrict__ X,
    float* __restrict__ acc, int nedges)
{
    const int e = (blockIdx.x * blockDim.x + threadIdx.x) >> 5;
    if (e >= nedges) return;
    const int lane = threadIdx.x & 31;

    const int r   = __builtin_nontemporal_load(rows + e);
    const int c   = __builtin_nontemporal_load(cols + e);
    const float v = __builtin_nontemporal_load(vals + e);

    const float4* __restrict__ src =
        (const float4*)(X + (size_t)c * DIM + lane * 8);
    float4 s0 = src[0];
    float4 s1 = src[1];

    float* __restrict__ dst = acc + (size_t)r * DIM + lane * 8;
    unsafeAtomicAdd(dst + 0, v * s0.x);
    unsafeAtomicAdd(dst + 1, v * s0.y);
    unsafeAtomicAdd(dst + 2, v * s0.z);
    unsafeAtomicAdd(dst + 3, v * s0.w);
    unsafeAtomicAdd(dst + 4, v * s1.x);
    unsafeAtomicAdd(dst + 5, v * s1.y);
    unsafeAtomicAdd(dst + 6, v * s1.z);
    unsafeAtomicAdd(dst + 7, v * s1.w);
}

// ---------------------------------------------------------------------------
// Gate: per row -> leaky, LayerNorm x3, attention dots, sigmoid, 3x3 mix,
// softmax/T, weighted combine. One wave per row; wave32 shuffle reductions.
// ---------------------------------------------------------------------------
__device__ __forceinline__ float wave_sum(float v) {
    #pragma unroll
    for (int off = 16; off > 0; off >>= 1)
        v += __shfl_xor(v, off, 32);
    return v;
}
__device__ __forceinline__ float leaky(float x) {
    return x >= 0.0f ? x : 0.01f * x;
}

__global__ __launch_bounds__(256) void gate_combine(
    const float* __restrict__ accL, const float* __restrict__ accH,
    const float* __restrict__ mlp,
    const float* __restrict__ gL, const float* __restrict__ bL,
    const float* __restrict__ gH, const float* __restrict__ bH,
    const float* __restrict__ gM, const float* __restrict__ bM,
    const float* __restrict__ avL, const float* __restrict__ avH,
    const float* __restrict__ avM, const float* __restrict__ av33,
    float* __restrict__ out, int nrows)
{
    const int row  = (blockIdx.x * blockDim.x + threadIdx.x) >> 5;
    if (row >= nrows) return;
    const int lane = threadIdx.x & 31;
    const int d0   = lane * 8;

    float lo[8], hi[8], ml[8];
    {
        const float4* pl = (const float4*)(accL + (size_t)row * DIM + d0);
        const float4* ph = (const float4*)(accH + (size_t)row * DIM + d0);
        const float4* pm = (const float4*)(mlp  + (size_t)row * DIM + d0);
        float4 a, b;
        a = pl[0]; b = pl[1];
        lo[0]=a.x; lo[1]=a.y; lo[2]=a.z; lo[3]=a.w; lo[4]=b.x; lo[5]=b.y; lo[6]=b.z; lo[7]=b.w;
        a = ph[0]; b = ph[1];
        hi[0]=a.x; hi[1]=a.y; hi[2]=a.z; hi[3]=a.w; hi[4]=b.x; hi[5]=b.y; hi[6]=b.z; hi[7]=b.w;
        a = pm[0]; b = pm[1];
        ml[0]=a.x; ml[1]=a.y; ml[2]=a.z; ml[3]=a.w; ml[4]=b.x; ml[5]=b.y; ml[6]=b.z; ml[7]=b.w;
    }
    #pragma unroll
    for (int i = 0; i < 8; ++i) { lo[i]=leaky(lo[i]); hi[i]=leaky(hi[i]); ml[i]=leaky(ml[i]); }

    float sl=0, ql=0, sh=0, qh=0, sm=0, qm=0;
    #pragma unroll
    for (int i = 0; i < 8; ++i) {
        sl += lo[i]; ql += lo[i]*lo[i];
        sh += hi[i]; qh += hi[i]*hi[i];
        sm += ml[i]; qm += ml[i]*ml[i];
    }
    sl = wave_sum(sl); ql = wave_sum(ql);
    sh = wave_sum(sh); qh = wave_sum(qh);
    sm = wave_sum(sm); qm = wave_sum(qm);

    const float inv = 1.0f / (float)DIM;
    const float mL = sl*inv, vL = ql*inv - mL*mL, rL = rsqrtf(vL + 1e-5f);
    const float mH = sh*inv, vH = qh*inv - mH*mH, rH = rsqrtf(vH + 1e-5f);
    const float mM = sm*inv, vM = qm*inv - mM*mM, rM = rsqrtf(vM + 1e-5f);

    float dL=0, dH=0, dM=0;
    #pragma unroll
    for (int i = 0; i < 8; ++i) {
        const int d = d0 + i;
        dL += ((lo[i]-mL)*rL*gL[d] + bL[d]) * avL[d];
        dH += ((hi[i]-mH)*rH*gH[d] + bH[d]) * avH[d];
        dM += ((ml[i]-mM)*rM*gM[d] + bM[d]) * avM[d];
    }
    dL = wave_sum(dL); dH = wave_sum(dH); dM = wave_sum(dM);

    const float sgL = 1.0f / (1.0f + __expf(-dL));
    const float sgH = 1.0f / (1.0f + __expf(-dH));
    const float sgM = 1.0f / (1.0f + __expf(-dM));

    float lg[3];
    #pragma unroll
    for (int j = 0; j < 3; ++j)
        lg[j] = (sgL*av33[0*3+j] + sgH*av33[1*3+j] + sgM*av33[2*3+j]) * (1.0f/3.0f);
    float mx = fmaxf(lg[0], fmaxf(lg[1], lg[2]));
    float e0 = __expf(lg[0]-mx), e1 = __expf(lg[1]-mx), e2 = __expf(lg[2]-mx);
    const float rs = 1.0f / (e0 + e1 + e2);
    const float a0 = 3.0f*e0*rs, a1 = 3.0f*e1*rs, a2 = 3.0f*e2*rs;

    float4 o0, o1;
    o0.x = a0*lo[0] + a1*hi[0] + a2*ml[0];
    o0.y = a0*lo[1] + a1*hi[1] + a2*ml[1];
    o0.z = a0*lo[2] + a1*hi[2] + a2*ml[2];
    o0.w = a0*lo[3] + a1*hi[3] + a2*ml[3];
    o1.x = a0*lo[4] + a1*hi[4] + a2*ml[4];
    o1.y = a0*lo[5] + a1*hi[5] + a2*ml[5];
    o1.z = a0*lo[6] + a1*hi[6] + a2*ml[6];
    o1.w = a0*lo[7] + a1*hi[7] + a2*ml[7];
    float4* po = (float4*)(out + (size_t)row * DIM + d0);
    po[0] = o0;
    po[1] = o1;
}

// ---------------------------------------------------------------------------
extern "C" void kernel_launch(void* const* d_in, const int* in_sizes, int n_in,
                              void* d_out, int out_size, void* d_ws, size_t ws_size,
                              hipStream_t stream)
{
    const float* x         = (const float*)d_in[0];
    const int*   low_rows  = (const int*)  d_in[1];
    const int*   low_cols  = (const int*)  d_in[2];
    const float* low_vals  = (const float*)d_in[3];
    const int*   high_rows = (const int*)  d_in[4];
    const int*   high_cols = (const int*)  d_in[5];
    const float* high_vals = (const float*)d_in[6];
    // d_in[7] = adj_low_unnormalized (unused)
    const float* w_low     = (const float*)d_in[8];
    const float* w_high    = (const float*)d_in[9];
    const float* w_mlp     = (const float*)d_in[10];
    const float* ln_low_g  = (const float*)d_in[11];
    const float* ln_low_b  = (const float*)d_in[12];
    const float* ln_high_g = (const float*)d_in[13];
    const float* ln_high_b = (const float*)d_in[14];
    const float* ln_mlp_g  = (const float*)d_in[15];
    const float* ln_mlp_b  = (const float*)d_in[16];
    const float* av_low    = (const float*)d_in[17];
    const float* av_high   = (const float*)d_in[18];
    const float* av_mlp    = (const float*)d_in[19];
    const float* av33      = (const float*)d_in[20];

    const int N = in_sizes[0] / DIM;
    const int E = in_sizes[1];

    float* ws0  = (float*)d_ws;                  // XW scratch, finally x@w_mlp
    float* accL = ws0  + (size_t)N * DIM;
    float* accH = accL + (size_t)N * DIM;

    (void)hipMemsetAsync(accL, 0, (size_t)2 * N * DIM * sizeof(float), stream);

    const int    gemm_blocks = (N + 63) / 64;           // 64 rows per block
    const size_t gemm_lds    = (size_t)(DIM * DIM + 64 * DIM) * sizeof(float); // 320KB
    const int    spmm_blocks = (E + 7) / 8;             // 1 wave per edge
    const int    gate_blocks = (N + 7) / 8;             // 1 wave per row

    // low branch
    gemm_wmma_f32<<<gemm_blocks, 256, gemm_lds, stream>>>(x, w_low, ws0, N);
    spmm_scatter<<<spmm_blocks, 256, 0, stream>>>(low_rows, low_cols, low_vals, ws0, accL, E);
    // high branch (ws0 reuse safe: same-stream ordering)
    gemm_wmma_f32<<<gemm_blocks, 256, gemm_lds, stream>>>(x, w_high, ws0, N);
    spmm_scatter<<<spmm_blocks, 256, 0, stream>>>(high_rows, high_cols, high_vals, ws0, accH, E);
    // mlp branch
    gemm_wmma_f32<<<gemm_blocks, 256, gemm_lds, stream>>>(x, w_mlp, ws0, N);

    gate_combine<<<gate_blocks, 256, 0, stream>>>(
        accL, accH, ws0,
        ln_low_g, ln_low_b, ln_high_g, ln_high_b, ln_mlp_g, ln_mlp_b,
        av_low, av_high, av_mlp, av33,
        (float*)d_out, N);
}